// CornerTree_10170482556963
// MI455X (gfx1250) — compile-verified
//
#include <hip/hip_runtime.h>
#include <hip/hip_bf16.h>

typedef float v2f __attribute__((ext_vector_type(2)));
typedef float v8f __attribute__((ext_vector_type(8)));

#define T_NEAR 0.05f
#define T_FAR  1.5f
#define NSAMP  128
#define BGCOL  1.0f
#define SIG_TH 1.0e-4f
#define MAX_CHILD 4680   /* (8^4-1)/7 internal nodes * 8 = 585*8 ints = 18720 B */

#ifndef __has_builtin
#define __has_builtin(x) 0
#endif
#if __has_builtin(__builtin_amdgcn_global_load_async_to_lds_b128)
#define HAVE_ASYNC_B128 1
#else
#define HAVE_ASYNC_B128 0
#endif

// Async b128 copy operates on 16-byte int-vector pointers:
// arg0: global (device AS) v4i*, arg1: LDS (AS3) v4i*, then imm offset, imm cpol.
typedef int v4i_raw __attribute__((__vector_size__(16)));
typedef __attribute__((address_space(1))) v4i_raw g_v4i;
typedef __attribute__((address_space(3))) v4i_raw l_v4i;

// A[ch][kk] for the SH contraction, block-diagonal over channels:
// interp[0..26] is coeffs laid out as (ch,k) with kk = ch*9 + k.
__device__ __forceinline__ float sh_A(int m, int kk, const float* sh) {
  return (m < 3 && kk < 27 && (kk / 9) == m) ? sh[kk % 9] : 0.0f;
}

__global__ __launch_bounds__(256) void plenoctree_render(
    const float* __restrict__ rays_o, const float* __restrict__ rays_d,
    const float* __restrict__ data, const int* __restrict__ child,
    const int* __restrict__ nids, const float* __restrict__ off3,
    const float* __restrict__ scl3, float* __restrict__ out,
    int nrays, int nchild)
{
  // ---- stage the (hot, serially-chased) child table into LDS ------------
  __shared__ int s_child[MAX_CHILD];
  {
    const int tid = (int)threadIdx.x;
    const int nb  = (int)blockDim.x;
#if HAVE_ASYNC_B128
    const int nchunk = nchild >> 2;                 // 16B per async b128
    for (int i = tid; i < nchunk; i += nb) {
      __builtin_amdgcn_global_load_async_to_lds_b128(
          (g_v4i*)(child + (size_t)i * 4),
          (l_v4i*)(&s_child[i * 4]), 0, 0);
    }
    for (int i = (nchunk << 2) + tid; i < nchild; i += nb)  // tail (none for 4680)
      s_child[i] = child[i];
#if __has_builtin(__builtin_amdgcn_s_wait_asynccnt)
    __builtin_amdgcn_s_wait_asynccnt(0);
#else
    asm volatile("s_wait_asynccnt 0x0" ::: "memory");
#endif
#else
    for (int i = tid; i < nchild; i += nb) s_child[i] = child[i];
#endif
  }
  __syncthreads();

  const int lane = threadIdx.x & 31;
  const int wv = (int)((blockIdx.x * blockDim.x + threadIdx.x) >> 5);
  if (wv >= nrays) return;  // wave-uniform: EXEC stays all-ones inside live waves

  const float ox = rays_o[wv*3+0], oy = rays_o[wv*3+1], oz = rays_o[wv*3+2];
  float dx = rays_d[wv*3+0], dy = rays_d[wv*3+1], dz = rays_d[wv*3+2];
  const float inv = rsqrtf(dx*dx + dy*dy + dz*dz);
  dx *= inv; dy *= inv; dz *= inv;

  const float fx = off3[0], fy = off3[1], fz = off3[2];
  const float sxx = scl3[0], syy = scl3[1], szz = scl3[2];

  // SH-9 basis of the normalized direction
  const float C0 = 0.28209479177387814f, C1 = 0.4886025119029199f;
  const float C20 = 1.0925484305920792f, C21 = -1.0925484305920792f;
  const float C22 = 0.31539156525252005f, C23 = -1.0925484305920792f;
  const float C24 = 0.5462742152960396f;
  float sh[9];
  sh[0] = C0;          sh[1] = -C1*dy;     sh[2] = C1*dz;
  sh[3] = -C1*dx;      sh[4] = C20*dx*dy;  sh[5] = C21*dy*dz;
  sh[6] = C22*(2.f*dz*dz - dx*dx - dy*dy);
  sh[7] = C23*dx*dz;   sh[8] = C24*(dx*dx - dy*dy);

  // Precompute A-matrix K-chunks (constant per wave).
  // 16x4 f32 A layout: reg0 = (lane<16 ? K=k0 : K=k0+2), reg1 = (K=k0+1 : K=k0+3), M = lane&15.
  float Aax[7], Aay[7];
  {
    const int m = lane & 15;
#pragma unroll
    for (int c7 = 0; c7 < 7; ++c7) {
      const int kk0 = c7 * 4;
      const int kx = (lane < 16) ? kk0     : kk0 + 2;
      const int ky = (lane < 16) ? kk0 + 1 : kk0 + 3;
      Aax[c7] = sh_A(m, kx, sh);
      Aay[c7] = sh_A(m, ky, sh);
    }
  }

  const float dt    = (T_FAR - T_NEAR) / (float)(NSAMP - 1);  // linspace step
  const float delta = (T_FAR - T_NEAR) / (float)NSAMP;        // tau step

  float accR = 0.f, accG = 0.f, accB = 0.f, accW = 0.f, carry = 0.f;

  for (int j = 0; j < 4; ++j) {
    const int s = j * 32 + lane;
    const float t = T_NEAR + (float)s * dt;
    float px = fx + (ox + dx * t) * sxx;
    float py = fy + (oy + dy * t) * syy;
    float pz = fz + (oz + dz * t) * szz;
    px = fminf(fmaxf(px, 0.f), 1.f - 1e-6f);
    py = fminf(fmaxf(py, 0.f), 1.f - 1e-6f);
    pz = fminf(fmaxf(pz, 0.f), 1.f - 1e-6f);

    // -------- octree descent out of LDS (branch-free, mirrors reference) --
    int node = 0; bool done = false;
    int lnode = 0, li = 0, lj = 0, lk = 0;
    float qx = px, qy = py, qz = pz;
    float posx = px, posy = py, posz = pz;
#pragma unroll
    for (int lvl = 0; lvl < 4; ++lvl) {
      int ii = (int)floorf(posx * 2.f); ii = ii < 0 ? 0 : (ii > 1 ? 1 : ii);
      int jj = (int)floorf(posy * 2.f); jj = jj < 0 ? 0 : (jj > 1 ? 1 : jj);
      int kk = (int)floorf(posz * 2.f); kk = kk < 0 ? 0 : (kk > 1 ? 1 : kk);
      const int c = s_child[node * 8 + ii * 4 + jj * 2 + kk];
      const float nx = posx * 2.f - (float)ii;
      const float ny = posy * 2.f - (float)jj;
      const float nz = posz * 2.f - (float)kk;
      const bool newly = (!done) && (c < 0);
      lnode = newly ? node : lnode;
      li = newly ? ii : li; lj = newly ? jj : lj; lk = newly ? kk : lk;
      qx = newly ? nx : qx; qy = newly ? ny : qy; qz = newly ? nz : qz;
      const bool cont = (!done) && (c >= 0);
      node = cont ? node + c : node;
      posx = cont ? nx : posx; posy = cont ? ny : posy; posz = cont ? nz : posz;
      done = done || newly;
    }

    // -------- gather 8 corners (112B rows, float4 vector loads) --------
    const int nbase = (lnode * 8 + li * 4 + lj * 2 + lk) * 8;
    const int4* nrow = reinterpret_cast<const int4*>(nids + nbase);  // 32B aligned
    const int4 n0 = nrow[0], n1 = nrow[1];
    const int cid[8] = {n0.x, n0.y, n0.z, n0.w, n1.x, n1.y, n1.z, n1.w};
#pragma unroll
    for (int c = 0; c < 8; ++c)
      __builtin_prefetch(data + (size_t)cid[c] * 28, 0, 1);  // global_prefetch_b8

    float interp[28];
#pragma unroll
    for (int q = 0; q < 28; ++q) interp[q] = 0.f;
#pragma unroll
    for (int c = 0; c < 8; ++c) {
      const float wx = (c & 4) ? qx : (1.f - qx);
      const float wy = (c & 2) ? qy : (1.f - qy);
      const float wz = (c & 1) ? qz : (1.f - qz);
      const float w = wx * wy * wz;
      const float4* row = reinterpret_cast<const float4*>(data + (size_t)cid[c] * 28);
#pragma unroll
      for (int q = 0; q < 7; ++q) {
        const float4 v = row[q];
        interp[q*4+0] = fmaf(w, v.x, interp[q*4+0]);
        interp[q*4+1] = fmaf(w, v.y, interp[q*4+1]);
        interp[q*4+2] = fmaf(w, v.z, interp[q*4+2]);
        interp[q*4+3] = fmaf(w, v.w, interp[q*4+3]);
      }
    }

    float sigma = interp[27];
    sigma = (sigma > SIG_TH) ? sigma : 0.f;
    const float tau = sigma * delta;

    // -------- WMMA SH contraction: D[ch][sample] = sum_kk A[ch][kk]*interp_sample[kk]
    // Tile0 columns = samples of lanes 0..15, Tile1 = samples of lanes 16..31.
    // B layout mirrors C/D striping: reg0 holds rows {k0,k0+2}, reg1 rows {k0+1,k0+3}.
    v8f d0 = {0.f,0.f,0.f,0.f,0.f,0.f,0.f,0.f};
    v8f d1 = {0.f,0.f,0.f,0.f,0.f,0.f,0.f,0.f};
#pragma unroll
    for (int c7 = 0; c7 < 7; ++c7) {
      const int kk0 = c7 * 4;
      const float x0 = __shfl_xor(interp[kk0+0], 16, 32);
      const float x1 = __shfl_xor(interp[kk0+1], 16, 32);
      const float x2 = __shfl_xor(interp[kk0+2], 16, 32);
      const float x3 = __shfl_xor(interp[kk0+3], 16, 32);
      v2f a;  a.x = Aax[c7];  a.y = Aay[c7];
      v2f b0; b0.x = (lane < 16) ? interp[kk0+0] : x2;
              b0.y = (lane < 16) ? interp[kk0+1] : x3;
      v2f b1; b1.x = (lane < 16) ? x0 : interp[kk0+2];
              b1.y = (lane < 16) ? x1 : interp[kk0+3];
      d0 = __builtin_amdgcn_wmma_f32_16x16x4_f32(false, a, false, b0, (short)0, d0, false, false);
      d1 = __builtin_amdgcn_wmma_f32_16x16x4_f32(false, a, false, b1, (short)0, d1, false, false);
    }
    // D layout: VGPR r, lanes 0-15 hold row r (channels 0..2 are rows 0..2), col = lane.
    const float rA = __shfl_xor(d1[0], 16, 32);
    const float gA = __shfl_xor(d1[1], 16, 32);
    const float bA = __shfl_xor(d1[2], 16, 32);
    const float rr = (lane < 16) ? d0[0] : rA;
    const float gg = (lane < 16) ? d0[1] : gA;
    const float bb = (lane < 16) ? d0[2] : bA;
    const float R = 1.f / (1.f + expf(-rr));
    const float G = 1.f / (1.f + expf(-gg));
    const float B = 1.f / (1.f + expf(-bb));

    // -------- transmittance: inclusive scan of tau across lanes + carry ----
    float incl = tau;
#pragma unroll
    for (int o = 1; o < 32; o <<= 1) {
      const float v = __shfl_up(incl, o, 32);
      incl += (lane >= o) ? v : 0.f;
    }
    const float excl  = incl - tau;
    const float Tr    = expf(-(carry + excl));
    const float alpha = 1.f - expf(-tau);
    const float w     = Tr * alpha;
    accR = fmaf(w, R, accR);
    accG = fmaf(w, G, accG);
    accB = fmaf(w, B, accB);
    accW += w;
    carry += __shfl(incl, 31, 32);
  }

  // -------- cross-lane reduction and output --------
#pragma unroll
  for (int o = 16; o > 0; o >>= 1) {
    accR += __shfl_xor(accR, o, 32);
    accG += __shfl_xor(accG, o, 32);
    accB += __shfl_xor(accB, o, 32);
    accW += __shfl_xor(accW, o, 32);
  }
  if (lane == 0) {
    const float bg = (1.f - accW) * BGCOL;
    out[wv*3+0] = accR + bg;
    out[wv*3+1] = accG + bg;
    out[wv*3+2] = accB + bg;
  }
}

extern "C" void kernel_launch(void* const* d_in, const int* in_sizes, int n_in,
                              void* d_out, int out_size, void* d_ws, size_t ws_size,
                              hipStream_t stream) {
  const float* rays_o = (const float*)d_in[0];
  const float* rays_d = (const float*)d_in[1];
  const float* data   = (const float*)d_in[2];
  const int*   child  = (const int*)d_in[3];
  const int*   nids   = (const int*)d_in[4];
  const float* off3   = (const float*)d_in[5];
  const float* scl3   = (const float*)d_in[6];
  float* out = (float*)d_out;

  const int nrays = in_sizes[0] / 3;          // 4096
  int nchild = in_sizes[3];                   // 4680 for TREE_DEPTH=4
  if (nchild > MAX_CHILD) nchild = MAX_CHILD; // safety clamp (fixed-depth tree)

  const int threads = 256;                    // 8 wave32 per block = 8 rays
  const int blocks = (nrays * 32 + threads - 1) / threads;
  plenoctree_render<<<blocks, threads, 0, stream>>>(
      rays_o, rays_d, data, child, nids, off3, scl3, out, nrays, nchild);
}